// GATEncoder_773094113575
// MI455X (gfx1250) — compile-verified
//
#include <hip/hip_runtime.h>
#include <hip/hip_bf16.h>

#define N_NODES  50000
#define IN_CH    128
#define CH       128          // HEADS*HID == OUT == 128 everywhere
#define NEG_SLOPE 0.2f

typedef float v2f __attribute__((ext_vector_type(2)));
typedef float v8f __attribute__((ext_vector_type(8)));

// ---------------------------------------------------------------------------
// fp32 WMMA GEMM: Out[nrows,128] = X[nrows,128] @ W[128,128]
// One wave -> one 16x16 output tile. 8 waves/block cover the 128 cols.
// nrows must be a multiple of 16 (50000 = 3125*16).
// A frag (16x4 f32): lanes 0-15 hold M=0..15 with K pair (k,k+1) in v[0],v[1];
// lanes 16-31 hold K pair (k+2,k+3).  B frag mirrors with N=lane%16.
// ---------------------------------------------------------------------------
__global__ __launch_bounds__(256) void gemm128_wmma_f32(
    const float* __restrict__ X, const float* __restrict__ W,
    float* __restrict__ Out) {
  const int wave = threadIdx.x >> 5;        // 0..7 -> column tile
  const int lane = threadIdx.x & 31;
  const int rt   = blockIdx.x;              // row tile
  const int m    = rt * 16 + (lane & 15);
  const int col  = wave * 16 + (lane & 15);
  const int koff = (lane >= 16) ? 2 : 0;

  const float* __restrict__ xrow = X + (size_t)m * IN_CH;
  v8f acc = {};
#pragma unroll 4
  for (int k = 0; k < IN_CH; k += 4) {
    v2f a, b;
    a[0] = xrow[k + koff];
    a[1] = xrow[k + koff + 1];
    b[0] = W[(size_t)(k + koff) * CH + col];
    b[1] = W[(size_t)(k + koff + 1) * CH + col];
    acc = __builtin_amdgcn_wmma_f32_16x16x4_f32(
        /*neg_a=*/false, a, /*neg_b=*/false, b,
        /*c_mod=*/(short)0, acc, /*reuse_a=*/false, /*reuse_b=*/false);
  }
  const int rbase = rt * 16 + ((lane >= 16) ? 8 : 0);
#pragma unroll
  for (int g = 0; g < 8; ++g)
    Out[(size_t)(rbase + g) * CH + col] = acc[g];
}

// ---------------------------------------------------------------------------
// Per-node attention dot products: al_s[n,h] = sum_c h[n,h*C+c]*a_src[h*C+c]
// float4-vectorized inner loop (C is 64 or 128, both /4).
// ---------------------------------------------------------------------------
__global__ void node_dots(const float* __restrict__ h,
                          const float* __restrict__ a_s,
                          const float* __restrict__ a_d,
                          float* __restrict__ al_s, float* __restrict__ al_d,
                          int nnodes, int heads) {
  int idx = blockIdx.x * blockDim.x + threadIdx.x;
  if (idx >= nnodes * heads) return;
  const int node = idx / heads;
  const int hh   = idx % heads;
  const int C    = CH / heads;
  const float4* row = (const float4*)(h + (size_t)node * CH + hh * C);
  const float4* as  = (const float4*)(a_s + hh * C);
  const float4* ad  = (const float4*)(a_d + hh * C);
  float s = 0.f, d = 0.f;
  for (int c = 0; c < C / 4; ++c) {
    float4 v = row[c], vs = as[c], vd = ad[c];
    s = fmaf(v.x, vs.x, fmaf(v.y, vs.y, fmaf(v.z, vs.z, fmaf(v.w, vs.w, s))));
    d = fmaf(v.x, vd.x, fmaf(v.y, vd.y, fmaf(v.z, vd.z, fmaf(v.w, vd.w, d))));
  }
  al_s[idx] = s;
  al_d[idx] = d;
}

__global__ void fill_f32(float* __restrict__ p, float v, int n) {
  int i = blockIdx.x * blockDim.x + threadIdx.x;
  if (i < n) p[i] = v;
}

__global__ void fill4_f32(float4* __restrict__ p, float v, int n4) {
  int i = blockIdx.x * blockDim.x + threadIdx.x;
  if (i < n4) p[i] = make_float4(v, v, v, v);
}

// Float atomic max via sign-split int/uint trick (init must be -inf).
__device__ inline void atomicMaxF(float* addr, float v) {
  if (v >= 0.f)
    atomicMax((int*)addr, __float_as_int(v));
  else
    atomicMin((unsigned int*)addr, __float_as_uint(v));
}

// Pass 1: edge logits (leaky relu) + segment max over dst.
__global__ void edge_logits(const int* __restrict__ ei,
                            const float* __restrict__ al_s,
                            const float* __restrict__ al_d,
                            float* __restrict__ ebuf, float* __restrict__ emax,
                            int E, int EE, int heads) {
  int idx = blockIdx.x * blockDim.x + threadIdx.x;
  if (idx >= EE * heads) return;
  const int e  = idx / heads;
  const int hh = idx - e * heads;
  const int src = (e < E) ? ei[e]     : (e - E);
  const int dst = (e < E) ? ei[E + e] : (e - E);
  float v = al_s[src * heads + hh] + al_d[dst * heads + hh];
  v = (v >= 0.f) ? v : NEG_SLOPE * v;
  ebuf[idx] = v;
  atomicMaxF(&emax[dst * heads + hh], v);
}

// Pass 2: ee = exp(e - max); denom[dst] += ee.
__global__ void edge_exp(const int* __restrict__ ei,
                         float* __restrict__ ebuf,
                         const float* __restrict__ emax,
                         float* __restrict__ denom,
                         int E, int EE, int heads) {
  int idx = blockIdx.x * blockDim.x + threadIdx.x;
  if (idx >= EE * heads) return;
  const int e  = idx / heads;
  const int hh = idx - e * heads;
  const int dst = (e < E) ? ei[E + e] : (e - E);
  float ee = __expf(ebuf[idx] - emax[dst * heads + hh]);
  ebuf[idx] = ee;
  atomicAdd(&denom[dst * heads + hh], ee);
}

// Pass 3: agg[dst, c..c+3] += alpha * h[src, c..c+3]
// 4 channels per thread: one b128 gather of h[src], alpha amortized 4x.
// A 4-channel group never crosses a head boundary (64 % 4 == 0).
__global__ void edge_agg(const int* __restrict__ ei,
                         const float* __restrict__ ebuf,
                         const float* __restrict__ denom,
                         const float* __restrict__ h,
                         float* __restrict__ agg,
                         int E, int EE, int heads) {
  long long idx = (long long)blockIdx.x * blockDim.x + threadIdx.x;
  if (idx >= (long long)EE * (CH / 4)) return;
  const int e = (int)(idx >> 5);            // CH/4 == 32 groups per edge
  const int c = ((int)idx & 31) << 2;       // channel base
  const int src = (e < E) ? ei[e]     : (e - E);
  const int dst = (e < E) ? ei[E + e] : (e - E);
  const int hh  = (heads == 2) ? (c >> 6) : 0;
  const float alpha =
      ebuf[e * heads + hh] / (denom[dst * heads + hh] + 1e-16f);
  const float4 hv = *(const float4*)(h + (size_t)src * CH + c);
  float* dp = agg + (size_t)dst * CH + c;
  atomicAdd(dp + 0, alpha * hv.x);
  atomicAdd(dp + 1, alpha * hv.y);
  atomicAdd(dp + 2, alpha * hv.z);
  atomicAdd(dp + 3, alpha * hv.w);
}

// bias add (+ optional ELU), in place over [nnodes,128], float4-vectorized
__global__ void bias_act(float4* __restrict__ p, const float4* __restrict__ b,
                         int n4, int do_elu) {
  int i = blockIdx.x * blockDim.x + threadIdx.x;
  if (i >= n4) return;
  float4 v = p[i];
  float4 bb = b[i & (CH / 4 - 1)];
  v.x += bb.x; v.y += bb.y; v.z += bb.z; v.w += bb.w;
  if (do_elu) {
    v.x = (v.x > 0.f) ? v.x : (__expf(v.x) - 1.f);
    v.y = (v.y > 0.f) ? v.y : (__expf(v.y) - 1.f);
    v.z = (v.z > 0.f) ? v.z : (__expf(v.z) - 1.f);
    v.w = (v.w > 0.f) ? v.w : (__expf(v.w) - 1.f);
  }
  p[i] = v;
}

static inline int cdiv(long long a, int b) { return (int)((a + b - 1) / b); }

extern "C" void kernel_launch(void* const* d_in, const int* in_sizes, int n_in,
                              void* d_out, int out_size, void* d_ws, size_t ws_size,
                              hipStream_t stream) {
  const float* x      = (const float*)d_in[0];
  const int*   ei     = (const int*)  d_in[1];
  const float* W1     = (const float*)d_in[2];
  const float* a_src1 = (const float*)d_in[3];
  const float* a_dst1 = (const float*)d_in[4];
  const float* b1     = (const float*)d_in[5];
  const float* W2     = (const float*)d_in[6];
  const float* a_src2 = (const float*)d_in[7];
  const float* a_dst2 = (const float*)d_in[8];
  const float* b2     = (const float*)d_in[9];
  float* out = (float*)d_out;

  const int N  = in_sizes[0] / IN_CH;   // 50000
  const int E  = in_sizes[1] / 2;       // 800000
  const int EE = E + N;                 // + self loops

  // workspace carve-up (floats)
  float* ws    = (float*)d_ws;
  float* hbuf  = ws;                    // N*128  (h1, then h2)
  float* xbuf  = hbuf  + (size_t)N * CH;// N*128  (agg1 -> elu -> layer2 input)
  float* al_s  = xbuf  + (size_t)N * CH;// N*2
  float* al_d  = al_s  + (size_t)N * 2;
  float* emax  = al_d  + (size_t)N * 2;
  float* denom = emax  + (size_t)N * 2;
  float* ebuf  = denom + (size_t)N * 2; // EE*2

  const int T = 256;
  const int row_tiles = N / 16;         // 3125

  // ---------------- Layer 1 (heads = 2) ----------------
  gemm128_wmma_f32<<<row_tiles, T, 0, stream>>>(x, W1, hbuf);
  node_dots<<<cdiv((long long)N * 2, T), T, 0, stream>>>(hbuf, a_src1, a_dst1,
                                                         al_s, al_d, N, 2);
  fill_f32<<<cdiv((long long)N * 2, T), T, 0, stream>>>(emax, -INFINITY, N * 2);
  fill_f32<<<cdiv((long long)N * 2, T), T, 0, stream>>>(denom, 0.f, N * 2);
  fill4_f32<<<cdiv((long long)N * CH / 4, T), T, 0, stream>>>((float4*)xbuf, 0.f,
                                                              N * CH / 4);
  edge_logits<<<cdiv((long long)EE * 2, T), T, 0, stream>>>(ei, al_s, al_d,
                                                            ebuf, emax, E, EE, 2);
  edge_exp<<<cdiv((long long)EE * 2, T), T, 0, stream>>>(ei, ebuf, emax, denom,
                                                         E, EE, 2);
  edge_agg<<<cdiv((long long)EE * (CH / 4), T), T, 0, stream>>>(
      ei, ebuf, denom, hbuf, xbuf, E, EE, 2);
  bias_act<<<cdiv((long long)N * CH / 4, T), T, 0, stream>>>(
      (float4*)xbuf, (const float4*)b1, N * CH / 4, 1);

  // ---------------- Layer 2 (heads = 1) ----------------
  gemm128_wmma_f32<<<row_tiles, T, 0, stream>>>(xbuf, W2, hbuf);
  node_dots<<<cdiv((long long)N, T), T, 0, stream>>>(hbuf, a_src2, a_dst2,
                                                     al_s, al_d, N, 1);
  fill_f32<<<cdiv((long long)N, T), T, 0, stream>>>(emax, -INFINITY, N);
  fill_f32<<<cdiv((long long)N, T), T, 0, stream>>>(denom, 0.f, N);
  fill4_f32<<<cdiv((long long)N * CH / 4, T), T, 0, stream>>>((float4*)out, 0.f,
                                                              N * CH / 4);
  edge_logits<<<cdiv((long long)EE, T), T, 0, stream>>>(ei, al_s, al_d, ebuf,
                                                        emax, E, EE, 1);
  edge_exp<<<cdiv((long long)EE, T), T, 0, stream>>>(ei, ebuf, emax, denom,
                                                     E, EE, 1);
  edge_agg<<<cdiv((long long)EE * (CH / 4), T), T, 0, stream>>>(
      ei, ebuf, denom, hbuf, out, E, EE, 1);
  bias_act<<<cdiv((long long)N * CH / 4, T), T, 0, stream>>>(
      (float4*)out, (const float4*)b2, N * CH / 4, 0);
}